// Spectralpooling_66700842107029
// MI455X (gfx1250) — compile-verified
//
#include <hip/hip_runtime.h>
#include <math.h>

// Problem constants (from reference)
#define B_    16
#define C_    64
#define T_    2048
#define OUT_  512
#define NEED_ 257          // OUT_/2 + 1 retained rfft bins
#define K2_   514          // 2*NEED_ (interleaved Re/Im columns)
#define K2P_  544          // padded to 17*32 for clean tiling (pad cols/rows are zero)

static __device__ __constant__ float kTwoPi = 6.283185307179586f;

typedef float v2f __attribute__((ext_vector_type(2)));
typedef float v8f __attribute__((ext_vector_type(8)));

// ---------------------------------------------------------------------------
// Kernel A: build batch-independent inverse basis W [K2P_ x OUT_]
//   W[2k,   n] =  w_k * cos(2*pi*k*n/512) / 512
//   W[2k+1, n] = -w_k * sin(2*pi*k*n/512) / 512
// rows with k >= NEED_ (padding) are zero.
// ---------------------------------------------------------------------------
__global__ void sp_build_w(float* __restrict__ W) {
    int idx = blockIdx.x * blockDim.x + threadIdx.x;
    if (idx >= K2P_ * OUT_) return;
    int row = idx / OUT_;          // interleaved 2k / 2k+1 index
    int n   = idx - row * OUT_;
    int k   = row >> 1;
    float val = 0.0f;
    if (k < NEED_) {
        float wk = (k == 0 || k == NEED_ - 1) ? 1.0f : 2.0f;
        // exact integer phase reduction: k*n < 2^24
        int im = (k * n) & (OUT_ - 1);             // (k*n) mod 512
        float ang = kTwoPi * (float)im * (1.0f / (float)OUT_);
        float s, c;
        __sincosf(ang, &s, &c);
        val = (row & 1) ? (-wk * s * (1.0f / (float)OUT_))
                        : ( wk * c * (1.0f / (float)OUT_));
    }
    W[idx] = val;
}

// ---------------------------------------------------------------------------
// Kernel B: per-batch GEMM1 with on-the-fly DFT basis:
//   Y[b] (64 x K2P_) = x[b] (64 x 2048)  @  F_b (2048 x K2P_)
//   F_b[t, 2k]   = cos(-2*pi*k*t/L) = cos(2*pi*((k*t) mod L)/L)   (t < L)
//   F_b[t, 2k+1] = sin(-2*pi*k*t/L) = -sin(2*pi*((k*t) mod L)/L)  (t < L)
// Block: 256 threads = 8 waves; wave w -> (mtile = w>>1, ntile = w&1).
// Block covers all 64 rows x 32 Y-columns; grid = (B_, K2P_/32).
// ---------------------------------------------------------------------------
#define KSTEP1 64
__global__ __launch_bounds__(256) void sp_gemm1(const float* __restrict__ x,
                                                const int*   __restrict__ length,
                                                float*       __restrict__ Y) {
    int b  = blockIdx.x;
    int nb = blockIdx.y;                       // Y column block: cols nb*32 .. +31
    int Lr = length[b];
    int L  = Lr < 1 ? 1 : Lr;
    float invL = 1.0f / (float)L;

    __shared__ float sA[C_][KSTEP1 + 1];       // x tile   64 x 64 (+pad)
    __shared__ float sF[KSTEP1][32 + 1];       // F tile   64 x 32 (+pad)

    int tid   = threadIdx.x;
    int lane  = tid & 31;
    int wave  = tid >> 5;
    int mtile = wave >> 1;                     // 0..3
    int ntile = wave & 1;                      // 0..1

    v8f acc = {0.f, 0.f, 0.f, 0.f, 0.f, 0.f, 0.f, 0.f};

    const float* xb = x + (size_t)b * C_ * T_;

    for (int kk = 0; kk < T_; kk += KSTEP1) {
        // Stage x tile: 64x64 elements, 16 per thread, coalesced on t.
        #pragma unroll
        for (int i = 0; i < (C_ * KSTEP1) / 256; ++i) {
            int li = tid + i * 256;
            int r  = li >> 6;                  // channel row
            int c  = li & (KSTEP1 - 1);        // t offset
            sA[r][c] = xb[r * T_ + kk + c];
        }
        // Synthesize F tile: 64x32 elements, 8 per thread.
        #pragma unroll
        for (int i = 0; i < (KSTEP1 * 32) / 256; ++i) {
            int li = tid + i * 256;
            int tr = li >> 5;                  // t row within tile
            int nc = li & 31;                  // col within tile
            int t  = kk + tr;
            int g2 = nb * 32 + nc;             // global interleaved column
            int k  = g2 >> 1;
            float v = 0.0f;
            if (t < L && k < NEED_) {
                int im = (k * t) % L;          // exact: k*t < 2^24
                float ang = kTwoPi * (float)im * invL;
                float s, c;
                __sincosf(ang, &s, &c);
                v = (g2 & 1) ? -s : c;
            }
            sF[tr][nc] = v;
        }
        __syncthreads();

        // WMMA fp32 16x16x4 over this K-chunk (16 MACs).
        int arow = mtile * 16 + (lane & 15);
        int bcol = ntile * 16 + (lane & 15);
        int koff = (lane >> 4) << 1;           // lanes 0-15: K 0,1; lanes 16-31: K 2,3
        #pragma unroll
        for (int ks = 0; ks < KSTEP1; ks += 4) {
            v2f a, bf;
            a.x  = sA[arow][ks + koff];
            a.y  = sA[arow][ks + koff + 1];
            bf.x = sF[ks + koff][bcol];
            bf.y = sF[ks + koff + 1][bcol];
            acc = __builtin_amdgcn_wmma_f32_16x16x4_f32(
                false, a, false, bf, (short)0, acc, false, false);
        }
        __syncthreads();
    }

    // Store Y tile: C/D layout — VGPR i: rows i (lanes 0-15) / i+8 (lanes 16-31).
    int colg    = nb * 32 + ntile * 16 + (lane & 15);
    int rowbase = mtile * 16 + ((lane >> 4) ? 8 : 0);
    float* Yb = Y + (size_t)b * C_ * K2P_;
    #pragma unroll
    for (int i = 0; i < 8; ++i) {
        Yb[(rowbase + i) * K2P_ + colg] = acc[i];
    }
}

// ---------------------------------------------------------------------------
// Kernel C: per-batch GEMM2 + branch-select epilogue:
//   pooled[b] (64 x 512) = Y[b] (64 x K2P_) @ W (K2P_ x 512)
//   out[b,c,n] = (L[b] < 512) ? (n < L[b] ? x[b,c,n] : 0) : pooled[b,c,n]
// ---------------------------------------------------------------------------
#define KSTEP2 32
__global__ __launch_bounds__(256) void sp_gemm2(const float* __restrict__ Y,
                                                const float* __restrict__ W,
                                                const float* __restrict__ x,
                                                const int*   __restrict__ length,
                                                float*       __restrict__ out) {
    int b  = blockIdx.x;
    int nb = blockIdx.y;                       // output cols nb*32 .. +31
    int L  = length[b];

    __shared__ float sY[C_][KSTEP2 + 1];       // 64 x 32 (+pad)
    __shared__ float sW[KSTEP2][32 + 1];       // 32 x 32 (+pad)

    int tid   = threadIdx.x;
    int lane  = tid & 31;
    int wave  = tid >> 5;
    int mtile = wave >> 1;
    int ntile = wave & 1;

    v8f acc = {0.f, 0.f, 0.f, 0.f, 0.f, 0.f, 0.f, 0.f};

    const float* Yb = Y + (size_t)b * C_ * K2P_;

    for (int kk = 0; kk < K2P_; kk += KSTEP2) {
        // Stage Y tile: 64x32, 8 per thread.
        #pragma unroll
        for (int i = 0; i < (C_ * KSTEP2) / 256; ++i) {
            int li = tid + i * 256;
            int r  = li >> 5;
            int c  = li & (KSTEP2 - 1);
            sY[r][c] = Yb[r * K2P_ + kk + c];
        }
        // Stage W tile: 32x32, 4 per thread.
        #pragma unroll
        for (int i = 0; i < (KSTEP2 * 32) / 256; ++i) {
            int li = tid + i * 256;
            int r  = li >> 5;
            int c  = li & 31;
            sW[r][c] = W[(kk + r) * OUT_ + nb * 32 + c];
        }
        __syncthreads();

        int arow = mtile * 16 + (lane & 15);
        int bcol = ntile * 16 + (lane & 15);
        int koff = (lane >> 4) << 1;
        #pragma unroll
        for (int ks = 0; ks < KSTEP2; ks += 4) {
            v2f a, bf;
            a.x  = sY[arow][ks + koff];
            a.y  = sY[arow][ks + koff + 1];
            bf.x = sW[ks + koff][bcol];
            bf.y = sW[ks + koff + 1][bcol];
            acc = __builtin_amdgcn_wmma_f32_16x16x4_f32(
                false, a, false, bf, (short)0, acc, false, false);
        }
        __syncthreads();
    }

    // Epilogue with short-sequence branch.
    int n       = nb * 32 + ntile * 16 + (lane & 15);   // output position 0..511
    int rowbase = mtile * 16 + ((lane >> 4) ? 8 : 0);
    const float* xb = x + (size_t)b * C_ * T_;
    float* ob = out + (size_t)b * C_ * OUT_;
    bool shortSeq = (L < OUT_);
    #pragma unroll
    for (int i = 0; i < 8; ++i) {
        int c = rowbase + i;
        float v = acc[i];
        if (shortSeq) v = (n < L) ? xb[c * T_ + n] : 0.0f;
        ob[c * OUT_ + n] = v;
    }
}

// ---------------------------------------------------------------------------
// Host launch. Workspace layout (floats):
//   W : K2P_ * OUT_            = 278,528
//   Y : B_ * C_ * K2P_         = 557,056
// total ~3.2 MB.
// ---------------------------------------------------------------------------
extern "C" void kernel_launch(void* const* d_in, const int* in_sizes, int n_in,
                              void* d_out, int out_size, void* d_ws, size_t ws_size,
                              hipStream_t stream) {
    (void)in_sizes; (void)n_in; (void)out_size; (void)ws_size;
    const float* x      = (const float*)d_in[0];
    const int*   length = (const int*)d_in[1];
    float*       out    = (float*)d_out;

    float* W = (float*)d_ws;
    float* Y = W + (size_t)K2P_ * OUT_;

    sp_build_w<<<(K2P_ * OUT_ + 255) / 256, 256, 0, stream>>>(W);
    sp_gemm1<<<dim3(B_, K2P_ / 32), 256, 0, stream>>>(x, length, Y);
    sp_gemm2<<<dim3(B_, OUT_ / 32), 256, 0, stream>>>(Y, W, x, length, out);
}